// YOLOForw_38208029066064
// MI455X (gfx1250) — compile-verified
//
#include <hip/hip_runtime.h>
#include <cstdint>

// ---------------------------------------------------------------------------
// YOLO decode head for gfx1250 (MI455X).
// Memory-bound transpose + elementwise transcendental; roofline ~10.6us
// (247 MB total traffic / 23.3 TB/s). No matrix math -> no WMMA; the CDNA5
// win here is the async LDS->global store engine (ASYNCcnt) + LDS transpose
// for fully coalesced reads AND writes, with NT hints on the streaming reads.
// ---------------------------------------------------------------------------

#define TPB        256
#define PIX_TILE   32          // pixels per block tile (one wave width)
#define CHANNELS   255         // 3 anchors * 85 attrs
#define TOTAL_ROWS 22743       // (76^2 + 38^2 + 19^2) * 3

__device__ __forceinline__ float fsigmoid(float x) {
    // 1 / (1 + 2^(-x*log2e))  -> v_exp_f32 + v_rcp_f32
    return __builtin_amdgcn_rcpf(1.0f + __builtin_amdgcn_exp2f(-1.4426950408889634f * x));
}
__device__ __forceinline__ float fexp(float x) {
    return __builtin_amdgcn_exp2f(1.4426950408889634f * x);
}

#if __has_builtin(__builtin_amdgcn_global_store_async_from_lds_b32)
#define YOLO_ASYNC_STORE 1
#else
#define YOLO_ASYNC_STORE 0
#endif

template <int W, int STRIDE>
__global__ __launch_bounds__(TPB)
void yolo_decode_kernel(const float* __restrict__ src,   // [bs, 255, W, W]
                        const float* __restrict__ idf,   // [80]
                        float* __restrict__ out,         // [bs, TOTAL_ROWS, 85]
                        int rowOff,
                        float aw0, float aw1, float aw2,
                        float ah0, float ah1, float ah2)
{
    constexpr int HW    = W * W;
    constexpr int TILES = (HW + PIX_TILE - 1) / PIX_TILE;

    __shared__ float tile[PIX_TILE * CHANNELS];   // 32640 B
    __shared__ float s_idf[80];

    const int b    = blockIdx.x / TILES;
    const int t    = blockIdx.x - b * TILES;
    const int p0   = t * PIX_TILE;
    const int valid = (HW - p0) < PIX_TILE ? (HW - p0) : PIX_TILE;

    if (threadIdx.x < 80) s_idf[threadIdx.x] = idf[threadIdx.x];
    __syncthreads();

    // ---------------- Phase 1: coalesced gather + decode into LDS ----------
    const int pix    = threadIdx.x & 31;   // lane -> pixel within tile
    const int chBase = threadIdx.x >> 5;   // wave  -> channel phase (0..7)
    const bool pOK   = pix < valid;
    const int  p     = p0 + (pOK ? pix : 0);          // clamp OOB lanes in-bounds
    const int  hi    = p / W;                          // compile-time W -> magic mul
    const int  wi    = p - hi * W;
    const float fh   = (float)hi;
    const float fw   = (float)wi;

    const float* inB = src + (size_t)b * (size_t)CHANNELS * (size_t)HW;

    int a = 0;            // chBase < 8 -> anchor 0, attr chBase
    int c = chBase;
    for (int ch = chBase; ch < CHANNELS; ch += 8) {
        float v = __builtin_nontemporal_load(&inB[(size_t)ch * HW + p]);
        float r;
        if (c == 0)      r = (fsigmoid(v) + fw) * (float)STRIDE;
        else if (c == 1) r = (fsigmoid(v) + fh) * (float)STRIDE;
        else if (c == 2) r = fexp(v) * ((a == 0) ? aw0 : (a == 1) ? aw1 : aw2);
        else if (c == 3) r = fexp(v) * ((a == 0) ? ah0 : (a == 1) ? ah1 : ah2);
        else if (c == 4) r = fsigmoid(v);
        else             r = fsigmoid(s_idf[c - 5] * v);
        // pix*255 + ch == output-contiguous order; stride 255 == bank-conflict-free
        tile[pix * CHANNELS + ch] = r;
        c += 8; if (c >= 85) { c -= 85; ++a; }
    }
    __syncthreads();

    // ---------------- Phase 2: contiguous, coalesced streaming store -------
    const int n = valid * CHANNELS;
    float* outP = out + (size_t)b * (size_t)TOTAL_ROWS * 85u
                      + ((size_t)rowOff + (size_t)p0 * 3u) * 85u;

#if YOLO_ASYNC_STORE
    for (int i = threadIdx.x; i < n; i += TPB) {
        __builtin_amdgcn_global_store_async_from_lds_b32(
            (__attribute__((address_space(1))) int*)(uintptr_t)(outP + i),
            (__attribute__((address_space(3))) int*)(uintptr_t)(&tile[i]),
            0, 0);
    }
#if __has_builtin(__builtin_amdgcn_s_wait_asynccnt)
    __builtin_amdgcn_s_wait_asynccnt(0);
#else
    asm volatile("s_wait_asynccnt 0" ::: "memory");
#endif
#else
    for (int i = threadIdx.x; i < n; i += TPB) {
        __builtin_nontemporal_store(tile[i], &outP[i]);
    }
#endif
}

extern "C" void kernel_launch(void* const* d_in, const int* in_sizes, int n_in,
                              void* d_out, int out_size, void* d_ws, size_t ws_size,
                              hipStream_t stream)
{
    const float* x0  = (const float*)d_in[0];   // [bs,255,76,76]
    const float* x1  = (const float*)d_in[1];   // [bs,255,38,38]
    const float* x2  = (const float*)d_in[2];   // [bs,255,19,19]
    const float* idf = (const float*)d_in[3];   // [80]
    float* out = (float*)d_out;

    const int bs = in_sizes[0] / (CHANNELS * 76 * 76);

    {   // scale 0: 76x76, stride 8, anchors in pixels
        constexpr int W = 76, HW = W * W, TILES = (HW + PIX_TILE - 1) / PIX_TILE;
        yolo_decode_kernel<76, 8><<<bs * TILES, TPB, 0, stream>>>(
            x0, idf, out, /*rowOff=*/0,
            10.0f, 16.0f, 33.0f, 13.0f, 30.0f, 23.0f);
    }
    {   // scale 1: 38x38, stride 16
        constexpr int W = 38, HW = W * W, TILES = (HW + PIX_TILE - 1) / PIX_TILE;
        yolo_decode_kernel<38, 16><<<bs * TILES, TPB, 0, stream>>>(
            x1, idf, out, /*rowOff=*/76 * 76 * 3,
            30.0f, 62.0f, 59.0f, 61.0f, 45.0f, 119.0f);
    }
    {   // scale 2: 19x19, stride 32
        constexpr int W = 19, HW = W * W, TILES = (HW + PIX_TILE - 1) / PIX_TILE;
        yolo_decode_kernel<19, 32><<<bs * TILES, TPB, 0, stream>>>(
            x2, idf, out, /*rowOff=*/76 * 76 * 3 + 38 * 38 * 3,
            116.0f, 156.0f, 373.0f, 90.0f, 198.0f, 326.0f);
    }
}